// DirectMap_28767690948680
// MI455X (gfx1250) — compile-verified
//
#include <hip/hip_runtime.h>
#include <hip/hip_bf16.h>
#include <math.h>

typedef __attribute__((ext_vector_type(2))) float v2f;
typedef __attribute__((ext_vector_type(8))) float v8f;

#define WAVES 4      // waves per workgroup
#define LDS_DW 4096  // 16 rows x 256 cols (fp32) per wave, fragment-blocked

// d_ws layout (float offsets)
#define OFF_VEC 0         // 46 tiles x 4096 DW, fragment-blocked dict embedding
#define OFF_CHOL 188416   // each repacked 256xN weight below
#define OFF_CW1 253952
#define OFF_CW2 319488
#define OFF_CW3 385024
#define OFF_CW4 450560    // 256x128
#define OFF_XW1 483328
#define OFF_XW2 548864
#define OFF_XW3 614400    // 256x128
// total = 647168 floats = 2,588,672 bytes

// ---------------------------------------------------------------------------
// fp32 WMMA: D(16x16) = A(16x4) * B(4x16) + C   (exact fp32 path on CDNA5)
// ---------------------------------------------------------------------------
__device__ __forceinline__ v8f wmma4(v2f a, v2f b, v8f c) {
  return __builtin_amdgcn_wmma_f32_16x16x4_f32(
      /*neg_a=*/false, a, /*neg_b=*/false, b,
      /*c_mod=*/(short)0, c, /*reuse_a=*/false, /*reuse_b=*/false);
}

// Fragment-blocked layout for a 16 x N fp32 tile: element (row, col) at DWORD
//   (col>>2)*64 + ((col&3)>>1)*32 + row*2 + (col&1)
// A-fragment kb = b64 at kb*64 + lane*2 (conflict-free / fully coalesced).
// The same layout serves as the B-fragment layout of the transposed operand.
__device__ __forceinline__ int blk_addr(int row, int col) {
  return (col >> 2) * 64 + ((col & 3) >> 1) * 32 + row * 2 + (col & 1);
}

__device__ __forceinline__ float half_max(float v) {
#pragma unroll
  for (int off = 1; off < 16; off <<= 1) v = fmaxf(v, __shfl_xor(v, off, 32));
  return v;
}
__device__ __forceinline__ float half_sum(float v) {
#pragma unroll
  for (int off = 1; off < 16; off <<= 1) v += __shfl_xor(v, off, 32);
  return v;
}

// Load 16xK activation tile as A fragments straight from global (row-major, ld)
template <int NKB>
__device__ __forceinline__ void load_af_global(v2f (&af)[64],
                                               const float* __restrict__ X,
                                               int ld, int lane) {
  const int r = lane & 15;
  const int cs = (lane >> 4) << 1;  // lanes 0-15: K+{0,1}; lanes 16-31: K+{2,3}
#pragma unroll
  for (int kb = 0; kb < NKB; ++kb)
    af[kb] = *(const v2f*)(X + r * ld + kb * 4 + cs);
}

// Load A fragments back from the blocked LDS buffer (conflict-free b64)
template <int NKB>
__device__ __forceinline__ void load_af_lds(v2f (&af)[64],
                                            const float* __restrict__ buf,
                                            int lane) {
#pragma unroll
  for (int kb = 0; kb < NKB; ++kb)
    af[kb] = *(const v2f*)(buf + kb * 64 + lane * 2);
}

// Linear(+bias)(+ReLU) with RAW row-major weights (2 scalar loads per WMMA).
// Only used by the tiny dictionary-embedding kernel.
template <int NKB, int N, bool RELU>
__device__ __forceinline__ void gemm_raw(const v2f (&af)[64],
                                         const float* __restrict__ W,
                                         const float* __restrict__ bias,
                                         float* __restrict__ buf, int lane) {
  const int r = lane & 15;
  const int hf = lane >> 4;
#pragma unroll 1
  for (int nt = 0; nt < (N >> 4); ++nt) {
    v8f ac0 = {0.f, 0.f, 0.f, 0.f, 0.f, 0.f, 0.f, 0.f};
    v8f ac1 = {0.f, 0.f, 0.f, 0.f, 0.f, 0.f, 0.f, 0.f};
    const float* wp = W + (hf * 2) * N + nt * 16 + r;
#pragma unroll
    for (int kb = 0; kb < NKB; kb += 2) {  // dual accumulators for ILP
      v2f b0, b1;
      b0.x = wp[(kb + 0) * 4 * N];
      b0.y = wp[(kb + 0) * 4 * N + N];
      b1.x = wp[(kb + 1) * 4 * N];
      b1.y = wp[(kb + 1) * 4 * N + N];
      ac0 = wmma4(af[kb + 0], b0, ac0);
      ac1 = wmma4(af[kb + 1], b1, ac1);
    }
    const float bv = bias[nt * 16 + r];
#pragma unroll
    for (int e = 0; e < 8; ++e) {
      float v = ac0[e] + ac1[e] + bv;
      if (RELU) v = fmaxf(v, 0.f);
      buf[blk_addr(e + hf * 8, nt * 16 + r)] = v;
    }
  }
}

// Linear(+bias)(+ReLU) with PACKED weights: one coalesced b64 per WMMA.
template <int NKB, int N, bool RELU>
__device__ __forceinline__ void gemm_packed(const v2f (&af)[64],
                                            const float* __restrict__ Wp,
                                            const float* __restrict__ bias,
                                            float* __restrict__ buf, int lane) {
  const int r = lane & 15;
  const int hf = lane >> 4;
#pragma unroll 1
  for (int nt = 0; nt < (N >> 4); ++nt) {
    v8f ac0 = {0.f, 0.f, 0.f, 0.f, 0.f, 0.f, 0.f, 0.f};
    v8f ac1 = {0.f, 0.f, 0.f, 0.f, 0.f, 0.f, 0.f, 0.f};
    const float* wp = Wp + (size_t)(nt * NKB) * 64 + lane * 2;
#pragma unroll
    for (int kb = 0; kb < NKB; kb += 2) {  // dual accumulators for ILP
      const v2f b0 = *(const v2f*)(wp + (kb + 0) * 64);  // 256B/wave each
      const v2f b1 = *(const v2f*)(wp + (kb + 1) * 64);
      ac0 = wmma4(af[kb + 0], b0, ac0);
      ac1 = wmma4(af[kb + 1], b1, ac1);
    }
    const float bv = bias[nt * 16 + r];
#pragma unroll
    for (int e = 0; e < 8; ++e) {
      float v = ac0[e] + ac1[e] + bv;
      if (RELU) v = fmaxf(v, 0.f);
      buf[blk_addr(e + hf * 8, nt * 16 + r)] = v;
    }
  }
}

// ---------------------------------------------------------------------------
// Kernel 0: repack the eight K=256 head weight matrices into fragment-blocked
// order so the GEMM B operand is a single coalesced b64 per WMMA.
//   Wp[(nt*64 + kb)*64 + lane*2 + c] = W[(4kb + 2*(lane>>4) + c)*N + nt*16 + (lane&15)]
// ---------------------------------------------------------------------------
struct RepackDesc { const float* src; int N; int dstOff; };
struct RepackParams { RepackDesc d[8]; float* ws; };

__global__ __launch_bounds__(256) void repack_kernel(RepackParams p) {
  const RepackDesc d = p.d[blockIdx.y];
  const int total = 256 * d.N;  // K == 256 for all repacked matrices
  float* dst = p.ws + d.dstOff;
  for (int o = blockIdx.x * 256 + threadIdx.x; o < total;
       o += gridDim.x * 256) {
    const int frag = o >> 6, q = o & 63;
    const int l = q >> 1, c = q & 1;
    const int kb = frag & 63, nt = frag >> 6;
    const int row = 4 * kb + ((l >> 4) << 1) + c;
    const int col = nt * 16 + (l & 15);
    dst[o] = d.src[row * d.N + col];
  }
}

// ---------------------------------------------------------------------------
// Kernel 1: vector = MLP(gau_dict) (730x3 -> 730x256), written to d_ws
// directly in fragment-blocked tiles (ready-to-use B fragments for softmax).
// ---------------------------------------------------------------------------
struct GdParams {
  const float* gau;
  const float *w1, *b1, *w2, *b2, *w3, *b3, *w4, *b4, *w5, *b5;
  float* vec;  // out: 46 tiles x 4096 DW, blocked
};

__global__ __launch_bounds__(WAVES * 32) void gd_vector_kernel(GdParams p) {
  __shared__ float lds[WAVES][LDS_DW];
  const int lane = threadIdx.x & 31;
  const int wv = threadIdx.x >> 5;
  const int tile = blockIdx.x * WAVES + wv;  // 16-row tile of the dictionary
  if (tile >= 46) return;                    // ceil(730/16)
  float* buf = lds[wv];
  const int r = lane & 15, hf = lane >> 4;
  const int row = tile * 16 + r;
  const int rc = row < 730 ? row : 729;  // pad rows 730-735 clamp to last real row 729

  v2f af[64];

  // ---- layer 1: 3 -> 128 (K padded to 4 with zeros), branchless loads ----
  const float* g3 = p.gau + rc * 3;
  v2f a0;
  {
    const float g0 = g3[0], g1 = g3[1], g2 = g3[2];
    a0.x = hf ? g2 : g0;
    a0.y = hf ? 0.f : g1;
  }
#pragma unroll 1
  for (int nt = 0; nt < 8; ++nt) {
    const int col = nt * 16 + r;
    const float wr = p.w1[(hf * 2) * 128 + col];  // row 0 or 2, in-bounds
    const float w1r = p.w1[1 * 128 + col];        // row 1
    v2f b;
    b.x = wr;
    b.y = hf ? 0.f : w1r;
    v8f acc = {0.f, 0.f, 0.f, 0.f, 0.f, 0.f, 0.f, 0.f};
    acc = wmma4(a0, b, acc);
    const float bv = p.b1[col];
#pragma unroll
    for (int e = 0; e < 8; ++e)
      buf[blk_addr(e + hf * 8, col)] = fmaxf(acc[e] + bv, 0.f);
  }

  load_af_lds<32>(af, buf, lane);
  gemm_raw<32, 256, true>(af, p.w2, p.b2, buf, lane);
  load_af_lds<64>(af, buf, lane);
  gemm_raw<64, 256, true>(af, p.w3, p.b3, buf, lane);
  load_af_lds<64>(af, buf, lane);
  gemm_raw<64, 256, true>(af, p.w4, p.b4, buf, lane);
  load_af_lds<64>(af, buf, lane);
  gemm_raw<64, 256, false>(af, p.w5, p.b5, buf, lane);

  // blocked LDS tile is already in B-fragment order: straight b64 copy-out
  float* vdst = p.vec + (size_t)tile * LDS_DW;
#pragma unroll 1
  for (int kb = 0; kb < 64; ++kb)
    *(v2f*)(vdst + kb * 64 + lane * 2) = *(const v2f*)(buf + kb * 64 + lane * 2);
}

// ---------------------------------------------------------------------------
// Kernel 2: fused per-token heads. One wave = one 16-token tile.
// ---------------------------------------------------------------------------
struct MainParams {
  const float* feat;            // (65536, 256)
  const float* gau;             // (730, 3)
  const float* chol_b;          // (256)
  const float* vecB;            // blocked dict embedding (46*4096 DW)
  const float *pw_chol;         // packed weights
  const float *pw_c1, *pw_c2, *pw_c3, *pw_c4;
  const float *pw_x1, *pw_x2, *pw_x3;
  const float *cb1, *cb2, *cb3, *cb4, *cw5, *cb5;  // cw5: raw 128x3
  const float *xb1, *xb2, *xb3, *xw4, *xb4;        // xw4: raw 128x2
  float *out_color, *out_chol, *out_off;
};

__global__ __launch_bounds__(WAVES * 32) void fused_heads_kernel(MainParams p) {
  __shared__ float lds[WAVES][LDS_DW];
  const int lane = threadIdx.x & 31;
  const int wv = threadIdx.x >> 5;
  float* buf = lds[wv];
  const int r = lane & 15, hf = lane >> 4;
  const int t0 = (blockIdx.x * WAVES + wv) * 16;  // first token of tile
  const float* X = p.feat + (size_t)t0 * 256;

  v2f af[64];

  // ===================== color head: 256->256->256->256->128->3 ============
  load_af_global<64>(af, X, 256, lane);
  gemm_packed<64, 256, true>(af, p.pw_c1, p.cb1, buf, lane);
  load_af_lds<64>(af, buf, lane);
  gemm_packed<64, 256, true>(af, p.pw_c2, p.cb2, buf, lane);
  load_af_lds<64>(af, buf, lane);
  gemm_packed<64, 256, true>(af, p.pw_c3, p.cb3, buf, lane);
  load_af_lds<64>(af, buf, lane);
  gemm_packed<64, 128, true>(af, p.pw_c4, p.cb4, buf, lane);
  {  // final 128 -> 3 (tiny: VALU dot, K split across lane halves)
    float s0 = 0.f, s1 = 0.f, s2 = 0.f;
#pragma unroll 4
    for (int kk = 0; kk < 64; ++kk) {
      const int k = hf * 64 + kk;
      const float h = buf[blk_addr(r, k)];
      s0 += h * p.cw5[k * 3 + 0];
      s1 += h * p.cw5[k * 3 + 1];
      s2 += h * p.cw5[k * 3 + 2];
    }
    s0 += __shfl_xor(s0, 16, 32);
    s1 += __shfl_xor(s1, 16, 32);
    s2 += __shfl_xor(s2, 16, 32);
    if (hf == 0) {
      float* o = p.out_color + (size_t)(t0 + r) * 3;
      o[0] = s0 + p.cb5[0]; o[1] = s1 + p.cb5[1]; o[2] = s2 + p.cb5[2];
    }
  }

  // ===================== offset head: 256->256->256->128->2, tanh ==========
  load_af_global<64>(af, X, 256, lane);
  gemm_packed<64, 256, true>(af, p.pw_x1, p.xb1, buf, lane);
  load_af_lds<64>(af, buf, lane);
  gemm_packed<64, 256, true>(af, p.pw_x2, p.xb2, buf, lane);
  load_af_lds<64>(af, buf, lane);
  gemm_packed<64, 128, true>(af, p.pw_x3, p.xb3, buf, lane);
  {
    float s0 = 0.f, s1 = 0.f;
#pragma unroll 4
    for (int kk = 0; kk < 64; ++kk) {
      const int k = hf * 64 + kk;
      const float h = buf[blk_addr(r, k)];
      s0 += h * p.xw4[k * 2 + 0];
      s1 += h * p.xw4[k * 2 + 1];
    }
    s0 += __shfl_xor(s0, 16, 32);
    s1 += __shfl_xor(s1, 16, 32);
    if (hf == 0) {
      float* o = p.out_off + (size_t)(t0 + r) * 2;
      o[0] = tanhf(s0 + p.xb4[0]);
      o[1] = tanhf(s1 + p.xb4[1]);
    }
  }

  // ============ cholesky head: lrelu -> linear -> online softmax @ dict ====
  load_af_global<64>(af, X, 256, lane);
#pragma unroll
  for (int kb = 0; kb < 64; ++kb) {  // LeakyReLU(0.01), elementwise on frags
    af[kb].x = af[kb].x >= 0.f ? af[kb].x : 0.01f * af[kb].x;
    af[kb].y = af[kb].y >= 0.f ? af[kb].y : 0.01f * af[kb].y;
  }
  gemm_packed<64, 256, false>(af, p.pw_chol, p.chol_b, buf, lane);
  load_af_lds<64>(af, buf, lane);  // cf fragments, reused for 46 logit tiles

  // Online softmax with PER-LANE partial accumulators: only the running max
  // needs a cross-lane reduction per tile; the exp-sum and the three
  // dictionary-weighted sums are reduced across lanes once at the end.
  float mrun[8], sp[8], p0[8], p1[8], p2[8];
#pragma unroll
  for (int e = 0; e < 8; ++e) {
    mrun[e] = -__builtin_inff();
    sp[e] = p0[e] = p1[e] = p2[e] = 0.f;
  }

#pragma unroll 1
  for (int nt = 0; nt < 46; ++nt) {  // 736-wide padded sweep over 730 rows
    const int d = nt * 16 + r;
    const bool valid = d < 730;
    const int dc = valid ? d : 729;
    const float* vb = p.vecB + (size_t)nt * LDS_DW + lane * 2;  // blocked B
    v8f ac0 = {0.f, 0.f, 0.f, 0.f, 0.f, 0.f, 0.f, 0.f};
    v8f ac1 = {0.f, 0.f, 0.f, 0.f, 0.f, 0.f, 0.f, 0.f};
#pragma unroll
    for (int kb = 0; kb < 64; kb += 2) {
      const v2f b0 = *(const v2f*)(vb + (kb + 0) * 64);  // coalesced 256B
      const v2f b1 = *(const v2f*)(vb + (kb + 1) * 64);
      ac0 = wmma4(af[kb + 0], b0, ac0);
      ac1 = wmma4(af[kb + 1], b1, ac1);
    }
    const float g0 = p.gau[dc * 3 + 0];
    const float g1 = p.gau[dc * 3 + 1];
    const float g2 = p.gau[dc * 3 + 2];
#pragma unroll
    for (int e = 0; e < 8; ++e) {  // rows e (lanes 0-15) and e+8 (lanes 16-31)
      const float v = valid ? (ac0[e] + ac1[e]) : -__builtin_inff();
      const float mn = fmaxf(mrun[e], half_max(v));  // uniform across half
      const float sc = __expf(mrun[e] - mn);         // first tile: exp(-inf)=0
      const float w = __expf(v - mn);                // invalid lanes: 0
      sp[e] = sp[e] * sc + w;
      p0[e] = fmaf(w, g0, p0[e] * sc);
      p1[e] = fmaf(w, g1, p1[e] * sc);
      p2[e] = fmaf(w, g2, p2[e] * sc);
      mrun[e] = mn;
    }
  }
#pragma unroll
  for (int e = 0; e < 8; ++e) {  // deferred cross-lane reductions + write
    const float s = half_sum(sp[e]);
    const float b0 = half_sum(p0[e]);
    const float b1 = half_sum(p1[e]);
    const float b2 = half_sum(p2[e]);
    if (r == 0) {  // lane 0 writes row e, lane 16 writes row e+8
      const float inv = 1.f / s;
      float* o = p.out_chol + (size_t)(t0 + e + hf * 8) * 3;
      o[0] = b0 * inv; o[1] = b1 * inv; o[2] = b2 * inv;
    }
  }
}

// ---------------------------------------------------------------------------
extern "C" void kernel_launch(void* const* d_in, const int* in_sizes, int n_in,
                              void* d_out, int out_size, void* d_ws,
                              size_t ws_size, hipStream_t stream) {
  (void)in_sizes; (void)n_in; (void)out_size; (void)ws_size;
  const float* const* in = (const float* const*)d_in;
  float* ws = (float*)d_ws;

  // setup_inputs order: feat, gau_dict, chol_w, chol_b,
  //   gd_params (5x w,b) = idx 4..13, color_params (5x w,b) = idx 14..23,
  //   xy_params (4x w,b) = idx 24..31, bs = idx 32
  RepackParams rp;
  rp.ws = ws;
  rp.d[0] = {in[2],  256, OFF_CHOL};   // chol_w
  rp.d[1] = {in[14], 256, OFF_CW1};
  rp.d[2] = {in[16], 256, OFF_CW2};
  rp.d[3] = {in[18], 256, OFF_CW3};
  rp.d[4] = {in[20], 128, OFF_CW4};
  rp.d[5] = {in[24], 256, OFF_XW1};
  rp.d[6] = {in[26], 256, OFF_XW2};
  rp.d[7] = {in[28], 128, OFF_XW3};

  GdParams g;
  g.gau = in[1];
  g.w1 = in[4];  g.b1 = in[5];
  g.w2 = in[6];  g.b2 = in[7];
  g.w3 = in[8];  g.b3 = in[9];
  g.w4 = in[10]; g.b4 = in[11];
  g.w5 = in[12]; g.b5 = in[13];
  g.vec = ws + OFF_VEC;

  MainParams m;
  m.feat = in[0];
  m.gau = in[1];
  m.chol_b = in[3];
  m.vecB = ws + OFF_VEC;
  m.pw_chol = ws + OFF_CHOL;
  m.pw_c1 = ws + OFF_CW1; m.pw_c2 = ws + OFF_CW2;
  m.pw_c3 = ws + OFF_CW3; m.pw_c4 = ws + OFF_CW4;
  m.pw_x1 = ws + OFF_XW1; m.pw_x2 = ws + OFF_XW2; m.pw_x3 = ws + OFF_XW3;
  m.cb1 = in[15]; m.cb2 = in[17]; m.cb3 = in[19]; m.cb4 = in[21];
  m.cw5 = in[22]; m.cb5 = in[23];
  m.xb1 = in[25]; m.xb2 = in[27]; m.xb3 = in[29];
  m.xw4 = in[30]; m.xb4 = in[31];

  float* out = (float*)d_out;
  const int n_tok = 16 * 256 * 16;       // 65536
  m.out_color = out;                     // (tok, 3)
  m.out_chol = out + (size_t)n_tok * 3;  // (tok, 3)
  m.out_off = out + (size_t)n_tok * 6;   // (tok, 2)

  // Kernel 0: weight repack (8 matrices, K=256)
  repack_kernel<<<dim3(64, 8), dim3(256), 0, stream>>>(rp);
  // Kernel 1: 46 dictionary tiles, 4 waves/block -> 12 blocks
  gd_vector_kernel<<<dim3(12), dim3(WAVES * 32), 0, stream>>>(g);
  // Kernel 2: 65536 tokens / 16 per wave / 4 waves/block -> 1024 blocks
  fused_heads_kernel<<<dim3(1024), dim3(WAVES * 32), 0, stream>>>(m);
}